// GCN_2_layers_28260884808301
// MI455X (gfx1250) — compile-verified
//
#include <hip/hip_runtime.h>
#include <hip/hip_bf16.h>

// GCN 4-layer pipeline for MI455X (gfx1250), wave32.
// Dense parts via V_WMMA_F32_16X16X4_F32 (exact fp32), B fragments register-
// resident across 5 M-tiles; edge propagation via coalesced float4 gathers +
// global_atomic_add_f32 scatters (L2-resident working set).

typedef __attribute__((ext_vector_type(2))) float v2f;
typedef __attribute__((ext_vector_type(8))) float v8f;

#define N_NODES 100000
#define N_EDGES 1600000
#define MT      5          // M-tiles per wave (6250 = 5 * 1250, exact)

__global__ void fill_kernel(float* __restrict__ p, int n, float v) {
    int i = blockIdx.x * blockDim.x + threadIdx.x;
    if (i < n) p[i] = v;
}

// narrow int64 edge indices to int32 (node ids < 100000)
__global__ void extract_kernel(const long long* __restrict__ row64,
                               const long long* __restrict__ col64,
                               int* __restrict__ rowi, int* __restrict__ coli, int e) {
    int i = blockIdx.x * blockDim.x + threadIdx.x;
    if (i < e) {
        rowi[i] = (int)row64[i];
        coli[i] = (int)col64[i];
    }
}

// degree of target nodes; float counts exact for deg << 2^24
__global__ void deg_kernel(const int* __restrict__ coli, float* __restrict__ deg, int e) {
    int i = blockIdx.x * blockDim.x + threadIdx.x;
    if (i < e) atomicAdd(&deg[coli[i]], 1.0f);
}

__global__ void dinv_kernel(float* __restrict__ deg, int n) {
    int i = blockIdx.x * blockDim.x + threadIdx.x;
    if (i < n) {
        float d = deg[i];
        deg[i] = (d > 0.0f) ? (1.0f / sqrtf(d)) : 0.0f;
    }
}

// per-edge coefficient: dinv[row]*dinv[col]*edge_attr  (reused by all 4 layers)
__global__ void wedge_kernel(const int* __restrict__ rowi, const int* __restrict__ coli,
                             const float* __restrict__ ea, const float* __restrict__ dinv,
                             float* __restrict__ w, int e) {
    int i = blockIdx.x * blockDim.x + threadIdx.x;
    if (i < e) w[i] = dinv[rowi[i]] * dinv[coli[i]] * ea[i];
}

// H = relu(A @ W); A:[N,64] row-major, W:[64,DOUT] row-major.
// One wave: one N-tile, MT consecutive M-tiles; B fragments preloaded once.
template <int DOUT>
__global__ void gemm_relu_wmma(const float* __restrict__ A, const float* __restrict__ W,
                               float* __restrict__ H, int total_waves) {
    int wave = (int)((blockIdx.x * blockDim.x + threadIdx.x) >> 5);
    if (wave >= total_waves) return;   // wave-uniform: EXEC stays all-1s for WMMA
    constexpr int NT = DOUT / 16;
    int lane = threadIdx.x & 31;
    int l16  = lane & 15;
    int khi  = (lane >> 4) << 1;               // 0 (lanes 0-15) or 2 (lanes 16-31)
    int mg   = wave / NT;                      // group of MT m-tiles
    int nt   = wave - mg * NT;
    int n    = nt * 16 + l16;

    // Preload all 16 B fragments (K=0..63) — register resident, reused MT times.
    const float* __restrict__ wp = W + (size_t)khi * DOUT + n;
    v2f b[16];
#pragma unroll
    for (int kk = 0; kk < 16; ++kk) {
        b[kk].x = wp[(size_t)(kk * 4) * DOUT];
        b[kk].y = wp[(size_t)(kk * 4 + 1) * DOUT];
    }

#pragma unroll
    for (int mt = 0; mt < MT; ++mt) {
        int mtile = mg * MT + mt;
        const float* __restrict__ arow = A + (size_t)(mtile * 16 + l16) * 64 + khi;
        v8f c = {};
#pragma unroll
        for (int kk = 0; kk < 16; ++kk) {
            v2f a;
            a.x = arow[kk * 4];
            a.y = arow[kk * 4 + 1];
            // (neg_a, A, neg_b, B, c_mod, C, reuse_a, reuse_b)
            c = __builtin_amdgcn_wmma_f32_16x16x4_f32(false, a, false, b[kk],
                                                      (short)0, c, false, false);
        }
        // C/D layout: VGPR v -> M = v + 8*(lane/16); N = lane%16
        float* __restrict__ hrow =
            H + (size_t)(mtile * 16 + ((lane >> 4) << 3)) * DOUT + n;
#pragma unroll
        for (int v = 0; v < 8; ++v) hrow[(size_t)v * DOUT] = fmaxf(c[v], 0.0f);
    }
}

// out[n, d] = b[d]  (bias folded into aggregation init)
__global__ void init_bias_kernel(float* __restrict__ out, const float* __restrict__ b,
                                 int total, int dmask) {
    int i = blockIdx.x * blockDim.x + threadIdx.x;
    if (i < total) out[i] = b[i & dmask];
}

// out[col[e], 4j..4j+3] += w[e] * h[row[e], 4j..4j+3]
template <int DOUT>
__global__ void scatter_kernel(const float* __restrict__ h,
                               const int* __restrict__ rowi, const int* __restrict__ coli,
                               const float* __restrict__ w, float* __restrict__ out,
                               int nedges) {
    constexpr int CH = DOUT / 4;   // float4 chunks per edge
    int t = blockIdx.x * blockDim.x + threadIdx.x;
    int e = t >> (DOUT == 64 ? 4 : 3);
    if (e >= nedges) return;
    int j = t & (CH - 1);
    int r    = rowi[e];
    int cdst = coli[e];
    float we = w[e];
    float4 v = ((const float4*)(h + (size_t)r * DOUT))[j];
    float* op = out + (size_t)cdst * DOUT + (size_t)j * 4;
    atomicAdd(op + 0, we * v.x);
    atomicAdd(op + 1, we * v.y);
    atomicAdd(op + 2, we * v.z);
    atomicAdd(op + 3, we * v.w);
}

__global__ void sigmoid_kernel(const float* __restrict__ in, float* __restrict__ out, int n) {
    int i = blockIdx.x * blockDim.x + threadIdx.x;
    if (i < n) out[i] = 1.0f / (1.0f + expf(-in[i]));
}

extern "C" void kernel_launch(void* const* d_in, const int* in_sizes, int n_in,
                              void* d_out, int out_size, void* d_ws, size_t ws_size,
                              hipStream_t stream) {
    (void)in_sizes; (void)n_in; (void)out_size; (void)ws_size;

    const float*     x     = (const float*)d_in[0];
    const long long* eidx  = (const long long*)d_in[1];   // int64 [2, E]
    const long long* row64 = eidx;                        // source (gather)
    const long long* col64 = eidx + N_EDGES;              // target (scatter)
    const float*     ea    = (const float*)d_in[2];

    const float* Ws[4] = { (const float*)d_in[3], (const float*)d_in[5],
                           (const float*)d_in[7], (const float*)d_in[9] };
    const float* Bs[4] = { (const float*)d_in[4], (const float*)d_in[6],
                           (const float*)d_in[8], (const float*)d_in[10] };
    float* out = (float*)d_out;

    // ws layout (64-elem aligned): dinv[N] | wedge[E] | rowi[E] | coli[E] | hbuf | abuf
    float* dinv  = (float*)d_ws;
    float* wedge = dinv + ((N_NODES + 63) & ~63);
    int*   rowi  = (int*)(wedge + N_EDGES);
    int*   coli  = rowi + N_EDGES;
    float* hbuf  = (float*)(coli + N_EDGES);
    float* abuf  = hbuf + (size_t)N_NODES * 64;

    const int B = 256;

    // --- edge prep + normalization coefficients (computed once) ---
    extract_kernel<<<(N_EDGES + B - 1) / B, B, 0, stream>>>(row64, col64, rowi, coli, N_EDGES);
    fill_kernel<<<(N_NODES + B - 1) / B, B, 0, stream>>>(dinv, N_NODES, 0.0f);
    deg_kernel<<<(N_EDGES + B - 1) / B, B, 0, stream>>>(coli, dinv, N_EDGES);
    dinv_kernel<<<(N_NODES + B - 1) / B, B, 0, stream>>>(dinv, N_NODES);
    wedge_kernel<<<(N_EDGES + B - 1) / B, B, 0, stream>>>(rowi, coli, ea, dinv, wedge, N_EDGES);

    // --- 4 GCN layers ---
    const float* in = x;
    for (int l = 0; l < 4; ++l) {
        int Dout = (l == 3) ? 32 : 64;
        int ntiles = Dout / 16;
        int total_waves = (N_NODES / 16 / MT) * ntiles;   // 1250 * NT (exact)
        int gblocks = (total_waves * 32 + B - 1) / B;
        if (Dout == 64)
            gemm_relu_wmma<64><<<gblocks, B, 0, stream>>>(in, Ws[l], hbuf, total_waves);
        else
            gemm_relu_wmma<32><<<gblocks, B, 0, stream>>>(in, Ws[l], hbuf, total_waves);

        int tot = N_NODES * Dout;
        init_bias_kernel<<<(tot + B - 1) / B, B, 0, stream>>>(abuf, Bs[l], tot, Dout - 1);

        int total = N_EDGES * (Dout / 4);                 // <= 25.6M, fits int
        int sblocks = (total + B - 1) / B;
        if (Dout == 64)
            scatter_kernel<64><<<sblocks, B, 0, stream>>>(hbuf, rowi, coli, wedge, abuf, N_EDGES);
        else
            scatter_kernel<32><<<sblocks, B, 0, stream>>>(hbuf, rowi, coli, wedge, abuf, N_EDGES);

        in = abuf;   // stream-ordered: safe to reuse as next layer's input
    }

    // --- final sigmoid ---
    int nout = N_NODES * 32;
    sigmoid_kernel<<<(nout + B - 1) / B, B, 0, stream>>>(abuf, out, nout);
}